// SubwordAggregation_3788161155116
// MI455X (gfx1250) — compile-verified
//
#include <hip/hip_runtime.h>
#include <hip/hip_bf16.h>
#include <stdint.h>

typedef float v2f __attribute__((ext_vector_type(2)));
typedef float v8f __attribute__((ext_vector_type(8)));

// ---------------------------------------------------------------------------
// Kernel 1: compact the True positions of a flat mask into src_list (rank -> index)
// Single workgroup, chunked block-wide inclusive scan.
// ---------------------------------------------------------------------------
__global__ __launch_bounds__(1024) void build_src_list(const uint8_t* __restrict__ mask,
                                                       int n,
                                                       int* __restrict__ src_list,
                                                       int* __restrict__ n_src) {
    __shared__ int sdata[1024];
    __shared__ int running;
    const int tid = threadIdx.x;
    const int nth = blockDim.x;
    if (tid == 0) running = 0;
    __syncthreads();
    for (int base = 0; base < n; base += nth) {
        const int i = base + tid;
        const int v = (i < n) ? (mask[i] != 0) : 0;
        sdata[tid] = v;
        __syncthreads();
        for (int off = 1; off < nth; off <<= 1) {
            const int t = (tid >= off) ? sdata[tid - off] : 0;
            __syncthreads();
            sdata[tid] += t;
            __syncthreads();
        }
        if (v) src_list[running + sdata[tid] - 1] = i;
        const int tot = sdata[nth - 1];
        __syncthreads();
        if (tid == 0) running += tot;
        __syncthreads();
    }
    if (tid == 0) *n_src = running;
}

// ---------------------------------------------------------------------------
// Kernel 2: from word_mask (B x maxW, prefix per row) and subword_mask
// (Nw x maxsub, prefix per row) build:
//   out_row[w]    = flat output slot (b*maxW + j) of word w
//   word_start[w] = exclusive prefix of per-word subword counts (global rank)
//   nw_out        = number of words Nw
// maxsub is derived on-device as sw_size / Nw.
// ---------------------------------------------------------------------------
__global__ __launch_bounds__(1024) void build_word_info(const uint8_t* __restrict__ word_mask,
                                                        int wm_size,
                                                        const uint8_t* __restrict__ sub_mask,
                                                        int sw_size,
                                                        int* __restrict__ out_row,
                                                        int* __restrict__ word_start,
                                                        int* __restrict__ nw_out) {
    __shared__ int sdata[1024];
    __shared__ int running;
    __shared__ int s_nw;
    const int tid = threadIdx.x;
    const int nth = blockDim.x;
    if (tid == 0) running = 0;
    __syncthreads();
    // Phase A: scan word_mask -> out_row, Nw
    for (int base = 0; base < wm_size; base += nth) {
        const int i = base + tid;
        const int v = (i < wm_size) ? (word_mask[i] != 0) : 0;
        sdata[tid] = v;
        __syncthreads();
        for (int off = 1; off < nth; off <<= 1) {
            const int t = (tid >= off) ? sdata[tid - off] : 0;
            __syncthreads();
            sdata[tid] += t;
            __syncthreads();
        }
        if (v) out_row[running + sdata[tid] - 1] = i;
        const int tot = sdata[nth - 1];
        __syncthreads();
        if (tid == 0) running += tot;
        __syncthreads();
    }
    if (tid == 0) { s_nw = running; *nw_out = running; running = 0; }
    __syncthreads();
    const int Nw = s_nw;
    const int maxsub = (Nw > 0) ? (sw_size / Nw) : 0;
    // Phase B: per-word subword counts (rows of sub_mask are prefix masks)
    for (int w = tid; w < Nw; w += nth) {
        int cnt = 0;
        for (int s = 0; s < maxsub; ++s) cnt += (sub_mask[w * maxsub + s] != 0);
        word_start[w] = cnt;
    }
    __syncthreads();
    // Phase C: exclusive scan of counts -> word_start, total in word_start[Nw]
    for (int base = 0; base < Nw; base += nth) {
        const int i = base + tid;
        const int v = (i < Nw) ? word_start[i] : 0;
        sdata[tid] = v;
        __syncthreads();
        for (int off = 1; off < nth; off <<= 1) {
            const int t = (tid >= off) ? sdata[tid - off] : 0;
            __syncthreads();
            sdata[tid] += t;
            __syncthreads();
        }
        if (i < Nw) word_start[i] = running + sdata[tid] - v;
        const int tot = sdata[nth - 1];
        __syncthreads();
        if (tid == 0) running += tot;
        __syncthreads();
    }
    if (tid == 0) word_start[Nw] = running;
}

// ---------------------------------------------------------------------------
// Kernel 3: per-word masked mean over gathered tokens. One block per word row
// (including pad rows >= Nw, which are zeroed so GEMM tiles stay clean).
// ---------------------------------------------------------------------------
__global__ __launch_bounds__(256) void mean_kernel(const float* __restrict__ inputs,
                                                   const int* __restrict__ src_list,
                                                   const int* __restrict__ word_start,
                                                   const int* __restrict__ nw_ptr,
                                                   float* __restrict__ means,
                                                   int H) {
    const int w = blockIdx.x;
    const int Nw = *nw_ptr;
    float* mrow = means + (size_t)w * H;
    if (w >= Nw) {
        for (int h = threadIdx.x; h < H; h += blockDim.x) mrow[h] = 0.0f;
        return;
    }
    const int s0 = word_start[w];
    const int s1 = word_start[w + 1];
    const float inv = 1.0f / (float)(s1 - s0);
    for (int h = threadIdx.x; h < H; h += blockDim.x) {
        float acc = 0.0f;
        for (int r = s0; r < s1; ++r) {
            const int tok = src_list[r];
            acc += inputs[(size_t)tok * H + h];
        }
        mrow[h] = acc * inv;
    }
}

// ---------------------------------------------------------------------------
// Kernel 4: D = tanh(means @ W^T + b), scattered through out_row.
// Block = 256 threads = 8 waves. Block covers one 16-row M tile and 8 adjacent
// 16-col N tiles; A tile staged in LDS in 16x64 chunks shared by all 8 waves.
// Each wave runs V_WMMA_F32_16X16X4_F32 over K.
// A layout (16x4 f32): lane m=lane&15, v0/v1 hold K = 2*(lane>>4) + {0,1}.
// B layout (4x16 f32): lane n=lane&15, v0/v1 hold K = 2*(lane>>4) + {0,1},
//   with B[k][n] = W[n][k] (W row-major, transposed product).
// C/D layout (16x16 f32): VGPR i -> M = i + 8*(lane>>4), N = lane&15.
// ---------------------------------------------------------------------------
__global__ __launch_bounds__(256) void gemm_tanh_scatter(const float* __restrict__ means,
                                                         const float* __restrict__ W,
                                                         const float* __restrict__ bias,
                                                         const int* __restrict__ out_row,
                                                         const int* __restrict__ nw_ptr,
                                                         float* __restrict__ out,
                                                         int H) {
    const int Nw    = *nw_ptr;
    const int mtile = blockIdx.y * 16;
    const int wave  = threadIdx.x >> 5;
    const int lane  = threadIdx.x & 31;
    const int ntile = (blockIdx.x * 8 + wave) * 16;
    const int nrow  = lane & 15;       // N (and M for A loads)
    const int hi    = lane >> 4;       // half-wave select
    const int kb    = hi * 2;          // K offset within group of 4

    __shared__ float lA[16][64 + 1];   // 16 rows x 64-K chunk of A (+1 pad)

    v8f acc = {0.f, 0.f, 0.f, 0.f, 0.f, 0.f, 0.f, 0.f};
    const float* Wrow = W + (size_t)(ntile + nrow) * H;   // B fragment source

    for (int k0 = 0; k0 < H; k0 += 64) {
        for (int idx = threadIdx.x; idx < 16 * 64; idx += 256) {
            const int r = idx >> 6, c = idx & 63;
            lA[r][c] = means[(size_t)(mtile + r) * H + k0 + c];
        }
        __syncthreads();
        for (int kk = 0; kk < 64; kk += 4) {
            v2f a = { lA[nrow][kk + kb], lA[nrow][kk + kb + 1] };
            v2f b = { Wrow[k0 + kk + kb], Wrow[k0 + kk + kb + 1] };
            acc = __builtin_amdgcn_wmma_f32_16x16x4_f32(
                /*neg_a=*/false, a, /*neg_b=*/false, b,
                /*c_mod=*/(short)0, acc, /*reuse_a=*/false, /*reuse_b=*/false);
        }
        __syncthreads();
    }

    const float bn = bias[ntile + nrow];
    for (int i = 0; i < 8; ++i) {
        const int m = mtile + i + 8 * hi;
        if (m < Nw) {
            const int row = out_row[m];
            out[(size_t)row * H + ntile + nrow] = tanhf(acc[i] + bn);
        }
    }
}

// ---------------------------------------------------------------------------
// Host launch. Input order (from setup_inputs dict):
//  0..2: question/table/column mask_plm       (bool, B*L each)
//  3..5: question/table/column subword masks  (bool, Nw*maxsub)
//  6..8: question/table/column word masks     (bool, B*maxW)
//  9: inputs (B,L,H) f32   10: W (H,H) f32    11: b (H,) f32
// Output: three (B, maxW, H) f32 blocks concatenated; chunk c size = in_sizes[6+c]*H.
// ---------------------------------------------------------------------------
extern "C" void kernel_launch(void* const* d_in, const int* in_sizes, int n_in,
                              void* d_out, int out_size, void* d_ws, size_t ws_size,
                              hipStream_t stream) {
    const int H = in_sizes[11];                       // 1024
    const float* inputs = (const float*)d_in[9];
    const float* W      = (const float*)d_in[10];
    const float* bias   = (const float*)d_in[11];
    float* out          = (float*)d_out;

    // Padding slots in the outputs must be exact zeros.
    hipMemsetAsync(d_out, 0, (size_t)out_size * sizeof(float), stream);

    char* ws = (char*)d_ws;
    size_t off = 0;
    auto alloc = [&](size_t bytes) -> void* {
        void* p = ws + off;
        off += (bytes + 255) & ~(size_t)255;
        return p;
    };

    size_t out_off = 0;
    for (int c = 0; c < 3; ++c) {
        const uint8_t* plm   = (const uint8_t*)d_in[c];
        const uint8_t* subm  = (const uint8_t*)d_in[3 + c];
        const uint8_t* wordm = (const uint8_t*)d_in[6 + c];
        const int plm_n = in_sizes[c];        // B*L
        const int sw_n  = in_sizes[3 + c];    // Nw * maxsub
        const int wm_n  = in_sizes[6 + c];    // B * maxW  (>= Nw)
        const int Mpad  = ((wm_n + 15) / 16) * 16;

        int*   src_list   = (int*)alloc((size_t)plm_n * sizeof(int));
        int*   n_src      = (int*)alloc(sizeof(int));
        int*   nw         = (int*)alloc(sizeof(int));
        int*   orow       = (int*)alloc((size_t)wm_n * sizeof(int));
        int*   word_start = (int*)alloc(((size_t)wm_n + 1) * sizeof(int));
        float* means      = (float*)alloc((size_t)Mpad * H * sizeof(float));

        build_src_list<<<1, 1024, 0, stream>>>(plm, plm_n, src_list, n_src);
        build_word_info<<<1, 1024, 0, stream>>>(wordm, wm_n, subm, sw_n,
                                                orow, word_start, nw);
        mean_kernel<<<Mpad, 256, 0, stream>>>(inputs, src_list, word_start, nw,
                                              means, H);
        gemm_tanh_scatter<<<dim3(H / 128, Mpad / 16), 256, 0, stream>>>(
            means, W, bias, orow, nw, out + out_off, H);

        out_off += (size_t)wm_n * H;
    }
    (void)n_in; (void)ws_size;
}